// canODE_attentionNoValue_64768106824056
// MI455X (gfx1250) — compile-verified
//
#include <hip/hip_runtime.h>
#include <hip/hip_bf16.h>

// ---------------------------------------------------------------------------
// canODE attention-no-value, MI455X (gfx1250, wave32).
// One workgroup (256 thr = 8 waves) per batch row. LDS-resident pipeline:
//   1) zero output rows, 2) stream-compact x row (order-preserving scan),
//   3) gather embed rows -> LDS, 4) qe/ke via V_WMMA_F32_16X16X4_F32
//      (two 256x64x64 f32 GEMMs, K padded 63->64 with a zero column),
//   5) 8 RK4 substeps using the rank-structure collapse:
//        fx_i = (v_i*wq0 + qe_i) . (S2*wk0 + m) / 8,  m = sum_j v_j*ke_j
//   6) scatter initial + final state into the 16MB output.
// All LDS 2D arrays padded (+1 row stride) to avoid 32-way bank conflicts.
// All reductions are fixed-order (shuffle + per-wave partials): deterministic.
// ---------------------------------------------------------------------------

typedef float v2f __attribute__((ext_vector_type(2)));
typedef float v8f __attribute__((ext_vector_type(8)));

#define NNZ    256
#define DDIM   8192
#define QK     64
#define NSUB   8

// LDS layout (float offsets)
#define EMB_S  65
#define KET_S  257
#define W_S    65
#define O_EMB   0
#define O_QE    (O_EMB + 256*EMB_S)      // 16640
#define O_KET   (O_QE  + 256*EMB_S)      // 33280
#define O_WQ    (O_KET + 64*KET_S)       // 49728
#define O_WK    (O_WQ  + 64*W_S)         // 53888
#define O_VAL   (O_WK  + 64*W_S)         // 58048
#define O_VBUF  (O_VAL  + 256)
#define O_RBUF  (O_VBUF + 256)
#define O_WQ0   (O_RBUF + 64)
#define O_WK0   (O_WQ0  + 64)
#define O_WPART (O_WK0  + 64)
#define O_SCAL  (O_WPART + 8)
#define O_POS   (O_SCAL  + 4)            // int region
#define O_SCAN  (O_POS   + 256)          // int region
#define SMEM_FLOATS (O_SCAN + 256)
#define SMEM_BYTES  (SMEM_FLOATS * 4)

__device__ __forceinline__ float wg_reduce_sum(float x, float* wpart, float* scal, int tid) {
    // fixed-order wave32 reduction, then 8 partials summed by thread 0
    #pragma unroll
    for (int o = 16; o > 0; o >>= 1) x += __shfl_down(x, o, 32);
    if ((tid & 31) == 0) wpart[tid >> 5] = x;
    __syncthreads();
    if (tid == 0) {
        float s = 0.f;
        #pragma unroll
        for (int w = 0; w < 8; ++w) s += wpart[w];
        scal[0] = s;
    }
    __syncthreads();
    return scal[0];
}

__device__ __forceinline__ float dxdt_eval(float vt, int tid,
                                           const float* qe, const float* keT,
                                           float* vbuf, float* rbuf,
                                           const float* wq0, const float* wk0,
                                           float* wpart, float* scal) {
    vbuf[tid] = vt;
    float S2 = wg_reduce_sum(vt * vt, wpart, scal, tid);   // barrier makes vbuf visible
    if (tid < QK) {
        const float* krow = keT + tid * KET_S;
        float m = 0.f;
        #pragma unroll 8
        for (int j = 0; j < NNZ; ++j) m += krow[j] * vbuf[j];
        rbuf[tid] = m + S2 * wk0[tid];                      // r = m + S2*wk0
    }
    __syncthreads();
    const float* qrow = qe + tid * EMB_S;
    float fx = 0.f;
    #pragma unroll
    for (int d = 0; d < QK; ++d) fx += (qrow[d] + vt * wq0[d]) * rbuf[d];
    fx *= 0.125f;                                           // QK^-0.5 = 1/8
    float s = wg_reduce_sum(vt * fx, wpart, scal, tid);
    return vt * (fx - s);
}

__global__ __launch_bounds__(256, 1)
void canode_kernel(const float* __restrict__ t, const float* __restrict__ x,
                   const float* __restrict__ embed, const float* __restrict__ Wq,
                   const float* __restrict__ bq, const float* __restrict__ Wk,
                   const float* __restrict__ bk, float* __restrict__ out) {
    extern __shared__ float sm[];
    float* emb  = sm + O_EMB;
    float* qe   = sm + O_QE;
    float* keT  = sm + O_KET;
    float* Wlq  = sm + O_WQ;
    float* Wlk  = sm + O_WK;
    float* valb = sm + O_VAL;
    float* vbuf = sm + O_VBUF;
    float* rbuf = sm + O_RBUF;
    float* wq0  = sm + O_WQ0;
    float* wk0  = sm + O_WK0;
    float* wpart= sm + O_WPART;
    float* scal = sm + O_SCAL;
    int*   posb = (int*)(sm + O_POS);
    int*   scan = (int*)(sm + O_SCAN);

    const int b   = blockIdx.x;
    const int tid = threadIdx.x;
    const float dt = (t[1] - t[0]) * (1.0f / NSUB);

    // ---- zero both output rows for this batch (harness doesn't re-poison) --
    {
        float4 z = make_float4(0.f, 0.f, 0.f, 0.f);
        float4* r0 = (float4*)(out + (size_t)b * DDIM);
        float4* r1 = (float4*)(out + (size_t)256 * DDIM + (size_t)b * DDIM);
        #pragma unroll
        for (int i = tid; i < DDIM / 4; i += 256) { r0[i] = z; r1[i] = z; }
    }

    // ---- order-preserving stream compaction of x[b, :] --------------------
    valb[tid] = 0.f; posb[tid] = 0;
    const float4* xr = (const float4*)(x + (size_t)b * DDIM) + tid * 8; // 32 floats/thread
    float4 xs[8];
    int cnt = 0;
    #pragma unroll
    for (int i = 0; i < 8; ++i) {
        xs[i] = xr[i];
        cnt += (xs[i].x != 0.f) + (xs[i].y != 0.f) + (xs[i].z != 0.f) + (xs[i].w != 0.f);
    }
    scan[tid] = cnt;
    __syncthreads();
    for (int off = 1; off < 256; off <<= 1) {          // Hillis-Steele inclusive
        int add = (tid >= off) ? scan[tid - off] : 0;
        __syncthreads();
        scan[tid] += add;
        __syncthreads();
    }
    int ofs = scan[tid] - cnt;                          // exclusive offset
    {
        const int base = tid * 32;
        #pragma unroll
        for (int i = 0; i < 8; ++i) {
            float v4[4] = { xs[i].x, xs[i].y, xs[i].z, xs[i].w };
            #pragma unroll
            for (int j = 0; j < 4; ++j) {
                if (v4[j] != 0.f && ofs < NNZ) {
                    valb[ofs] = v4[j];
                    posb[ofs] = base + i * 4 + j + 1;   // pos = idx+1
                    ++ofs;
                }
            }
        }
    }

    // ---- stage Wq/Wk (padded) + first columns -----------------------------
    for (int i = tid; i < QK * QK; i += 256) {
        int r = i >> 6, c = i & 63;
        Wlq[r * W_S + c] = Wq[i];
        Wlk[r * W_S + c] = Wk[i];
    }
    if (tid < QK) { wq0[tid] = Wq[tid * QK]; wk0[tid] = Wk[tid * QK]; }
    __syncthreads();

    // ---- gather embed rows: emb[i][0]=0 (K-pad), emb[i][1+e]=table[pos_i][e]
    {
        const int p = posb[tid];
        const float* er = embed + (size_t)p * 63;
        float* dst = emb + tid * EMB_S;
        dst[0] = 0.f;
        #pragma unroll 7
        for (int e = 0; e < 63; ++e) dst[1 + e] = er[e];
    }
    __syncthreads();

    // ---- qe / keT via V_WMMA_F32_16X16X4_F32 ------------------------------
    // C(256x64) = A(256x64) * B(64x64),  B[k][n] = W[n][k]  (C = A @ W^T)
    // 2 GEMMs x 64 tiles = 128 tile-jobs, 16 per wave, 16 K-steps each.
    {
        const int wid   = tid >> 5;
        const int lane  = tid & 31;
        const int lhalf = lane >> 4;    // K sub-pair select
        const int lsub  = lane & 15;
        for (int job = wid * 16; job < wid * 16 + 16; ++job) {
            const int g    = job >> 6;       // 0: q, 1: k
            const int tile = job & 63;
            const int nb   = tile >> 4;      // 0..3
            const int mb   = tile & 15;      // 0..15
            const float* Wl = g ? Wlk : Wlq;
            const int M = mb * 16 + lsub;
            const int N = nb * 16 + lsub;
            v8f acc = {0.f, 0.f, 0.f, 0.f, 0.f, 0.f, 0.f, 0.f};
            #pragma unroll
            for (int kb = 0; kb < 16; ++kb) {
                const int kk = kb * 4 + 2 * lhalf;
                v2f a = { emb[M * EMB_S + kk], emb[M * EMB_S + kk + 1] };
                v2f bb = { Wl[N * W_S + kk], Wl[N * W_S + kk + 1] };
                acc = __builtin_amdgcn_wmma_f32_16x16x4_f32(
                        false, a, false, bb, (short)0, acc, false, false);
            }
            if (g == 0) {
                const float bias = bq[N];
                #pragma unroll
                for (int r = 0; r < 8; ++r)
                    qe[(mb * 16 + r + 8 * lhalf) * EMB_S + N] = acc[r] + bias;
            } else {
                const float bias = bk[N];
                #pragma unroll
                for (int r = 0; r < 8; ++r)
                    keT[N * KET_S + (mb * 16 + r + 8 * lhalf)] = acc[r] + bias;
            }
        }
    }
    __syncthreads();

    // ---- RK4, 8 substeps --------------------------------------------------
    float v = valb[tid];
    for (int s = 0; s < NSUB; ++s) {
        float k1 = dxdt_eval(v,                  tid, qe, keT, vbuf, rbuf, wq0, wk0, wpart, scal);
        float k2 = dxdt_eval(v + 0.5f * dt * k1, tid, qe, keT, vbuf, rbuf, wq0, wk0, wpart, scal);
        float k3 = dxdt_eval(v + 0.5f * dt * k2, tid, qe, keT, vbuf, rbuf, wq0, wk0, wpart, scal);
        float k4 = dxdt_eval(v + dt * k3,        tid, qe, keT, vbuf, rbuf, wq0, wk0, wpart, scal);
        v += dt * (1.0f / 6.0f) * (k1 + 2.f * k2 + 2.f * k3 + k4);
    }

    // ---- scatter initial (t=0) and final (t=1) states ---------------------
    {
        const int p = posb[tid];
        if (p > 0) {
            const int idx = p - 1;
            out[(size_t)b * DDIM + idx] = valb[tid];
            out[(size_t)256 * DDIM + (size_t)b * DDIM + idx] = v;
        }
    }
}

extern "C" void kernel_launch(void* const* d_in, const int* in_sizes, int n_in,
                              void* d_out, int out_size, void* d_ws, size_t ws_size,
                              hipStream_t stream) {
    (void)in_sizes; (void)n_in; (void)out_size; (void)d_ws; (void)ws_size;
    const float* t     = (const float*)d_in[0];
    const float* x     = (const float*)d_in[1];
    const float* embed = (const float*)d_in[2];
    const float* Wq    = (const float*)d_in[3];
    const float* bq    = (const float*)d_in[4];
    const float* Wk    = (const float*)d_in[5];
    const float* bk    = (const float*)d_in[6];
    float* out = (float*)d_out;

    canode_kernel<<<dim3(256), dim3(256), SMEM_BYTES, stream>>>(
        t, x, embed, Wq, bq, Wk, bk, out);
}